// LightGCN_86131274154844
// MI455X (gfx1250) — compile-verified
//
#include <hip/hip_runtime.h>
#include <hip/hip_bf16.h>
#include <stdint.h>

#define NUM_USERS 500000
#define NUM_ITEMS 300000
#define NUM_NODES (NUM_USERS + NUM_ITEMS)
#define NUM_EDGES 1250000
#define DIM 64
#define N_LAYERS 3
#define EPB 16   // edges staged per block tile (16 rows * 256B = 4KB LDS)

#if __has_builtin(__builtin_amdgcn_global_load_async_to_lds_b128)
#define HAVE_ASYNC_LDS 1
#else
#define HAVE_ASYNC_LDS 0
#endif

typedef __attribute__((ext_vector_type(4))) int v4i;
typedef __attribute__((address_space(1))) v4i* gv4i_p;   // global int4*
typedef __attribute__((address_space(3))) v4i* lv4i_p;   // LDS int4*

__device__ __forceinline__ void atomic_add_f32(float* p, float v) {
    (void)__hip_atomic_fetch_add(p, v, __ATOMIC_RELAXED, __HIP_MEMORY_SCOPE_AGENT);
}

__device__ __forceinline__ void wait_async_zero() {
#if __has_builtin(__builtin_amdgcn_s_wait_asynccnt)
    __builtin_amdgcn_s_wait_asynccnt(0);
#else
    asm volatile("s_wait_asynccnt 0" ::: "memory");
#endif
}

// ---------------- probe: FIRST kernel in module so the disasm snippet shows the
// exact lowering of our f32 atomic (want: global_atomic_add_f32, not a CAS loop).
// Never launched by kernel_launch. ----------------

__global__ void k_probe_atomic(float* p, float v) {
    atomic_add_f32(&p[threadIdx.x], v);
}

// ---------------- scatter: agg[dst] += h[src] (h pre-scaled by src_norm) ----------------
// One block of 256 threads handles EPB=16 edges: stage 16 rows (4KB) into LDS via
// async global->LDS B128 (one per thread), then 4 coalesced f32 atomics per thread.

__global__ void k_scatter(const float* __restrict__ h, const int* __restrict__ src,
                          const int* __restrict__ dst, float* __restrict__ agg) {
    __shared__ float smem[EPB * DIM];   // smem[e*64 + d]
    __shared__ int   sdst[EPB];
    const int t     = threadIdx.x;      // 0..255
    const int ebase = blockIdx.x * EPB; // NUM_EDGES/EPB exact (78125 tiles)

    if (t < EPB) sdst[t] = dst[ebase + t];

    {
        const int e = t >> 4;           // edge within tile
        const int c = t & 15;           // float4 chunk within row
        const int s = src[ebase + e];
        const float* gp = h + (size_t)s * DIM + (size_t)c * 4;
#if HAVE_ASYNC_LDS
        __builtin_amdgcn_global_load_async_to_lds_b128(
            (gv4i_p)(void*)gp, (lv4i_p)(void*)&smem[t * 4], 0, 0);
        wait_async_zero();
#else
        *(float4*)&smem[t * 4] = *(const float4*)gp;
#endif
    }
    __syncthreads();

#pragma unroll
    for (int k = 0; k < 4; ++k) {
        const int idx = k * 256 + t;    // 0..1023 == e*64 + d
        const int e = idx >> 6;
        const int d = idx & 63;
        atomic_add_f32(&agg[(size_t)sdst[e] * DIM + d], smem[idx]);
    }
}

// ---------------- degree / norm ----------------

__global__ void k_zero_u32(uint32_t* __restrict__ p, int n) {
    int i = blockIdx.x * blockDim.x + threadIdx.x;
    if (i < n) p[i] = 0u;
}

__global__ void k_count_deg(const int* __restrict__ src, const int* __restrict__ dst,
                            uint32_t* __restrict__ deg_out, uint32_t* __restrict__ deg_in) {
    int e = blockIdx.x * blockDim.x + threadIdx.x;
    if (e < NUM_EDGES) {
        atomicAdd(&deg_out[src[e]], 1u);
        atomicAdd(&deg_in[dst[e]], 1u);
    }
}

// In-place: read u32 count, write float norm = 1/sqrt(max(c,1))
__global__ void k_norm(uint32_t* __restrict__ deg, float* norm, int n) {
    int i = blockIdx.x * blockDim.x + threadIdx.x;
    if (i < n) {
        uint32_t c = deg[i];
        float d = (float)(c ? c : 1u);
        norm[i] = 1.0f / sqrtf(d);
    }
}

// ---------------- init: h = h0 * src_norm, out = 0.25*h0 ----------------

__global__ void k_init(const float4* __restrict__ ue, const float4* __restrict__ ie,
                       const float* __restrict__ src_norm,
                       float4* __restrict__ h, float4* __restrict__ out) {
    int i = blockIdx.x * blockDim.x + threadIdx.x;     // float4 index
    const int total = NUM_NODES * (DIM / 4);
    if (i >= total) return;
    int node = i >> 4;                                  // DIM/4 == 16
    float4 v = (node < NUM_USERS) ? ue[i] : ie[i - NUM_USERS * (DIM / 4)];
    float sn = src_norm[node];
    h[i]   = make_float4(v.x * sn, v.y * sn, v.z * sn, v.w * sn);
    out[i] = make_float4(0.25f * v.x, 0.25f * v.y, 0.25f * v.z, 0.25f * v.w);
}

__global__ void k_zero_f4(float4* __restrict__ p, int n) {
    int i = blockIdx.x * blockDim.x + threadIdx.x;
    if (i < n) p[i] = make_float4(0.f, 0.f, 0.f, 0.f);
}

// ---------------- scale + accumulate: val = agg*dst_norm; out += 0.25*val;
//                  agg := val*src_norm (in place -> becomes next layer's h) ----------------

__global__ void k_scale(float4* __restrict__ agg, float4* __restrict__ out,
                        const float* __restrict__ dst_norm,
                        const float* __restrict__ src_norm, int write_next) {
    int i = blockIdx.x * blockDim.x + threadIdx.x;
    const int total = NUM_NODES * (DIM / 4);
    if (i >= total) return;
    int node = i >> 4;
    float dn = dst_norm[node];
    float4 a = agg[i];
    float4 val = make_float4(a.x * dn, a.y * dn, a.z * dn, a.w * dn);
    float4 o = out[i];
    o.x += 0.25f * val.x; o.y += 0.25f * val.y;
    o.z += 0.25f * val.z; o.w += 0.25f * val.w;
    out[i] = o;
    if (write_next) {
        float sn = src_norm[node];
        agg[i] = make_float4(val.x * sn, val.y * sn, val.z * sn, val.w * sn);
    }
}

// ---------------- host ----------------

extern "C" void kernel_launch(void* const* d_in, const int* in_sizes, int n_in,
                              void* d_out, int out_size, void* d_ws, size_t ws_size,
                              hipStream_t stream) {
    (void)in_sizes; (void)n_in; (void)out_size; (void)ws_size;
    const float* ue  = (const float*)d_in[0];
    const float* ie  = (const float*)d_in[1];
    const int*   src = (const int*)d_in[2];
    const int*   dst = (const int*)d_in[3];
    float*       out = (float*)d_out;

    // Workspace layout (all offsets 256B-aligned):
    //   [0)                 : 2*NUM_NODES u32 degree counts, overwritten in place with f32 norms
    //   [6.4MB)             : hA  (NUM_NODES*DIM f32)
    //   [6.4MB + 204.8MB)   : hB  (NUM_NODES*DIM f32)
    char* ws = (char*)d_ws;
    uint32_t* deg      = (uint32_t*)ws;
    float*    norms    = (float*)ws;            // aliases deg after k_norm
    float*    src_norm = norms;
    float*    dst_norm = norms + NUM_NODES;
    float*    hA = (float*)(ws + (size_t)2 * NUM_NODES * sizeof(uint32_t));
    float*    hB = hA + (size_t)NUM_NODES * DIM;

    const int T = 256;
    const int n2     = 2 * NUM_NODES;
    const int total4 = NUM_NODES * (DIM / 4);

    k_zero_u32<<<(n2 + T - 1) / T, T, 0, stream>>>(deg, n2);
    k_count_deg<<<(NUM_EDGES + T - 1) / T, T, 0, stream>>>(src, dst, deg, deg + NUM_NODES);
    k_norm<<<(n2 + T - 1) / T, T, 0, stream>>>(deg, norms, n2);

    k_init<<<(total4 + T - 1) / T, T, 0, stream>>>(
        (const float4*)ue, (const float4*)ie, src_norm, (float4*)hA, (float4*)out);

    float* h   = hA;
    float* agg = hB;
    for (int l = 0; l < N_LAYERS; ++l) {
        k_zero_f4<<<(total4 + T - 1) / T, T, 0, stream>>>((float4*)agg, total4);
        k_scatter<<<NUM_EDGES / EPB, T, 0, stream>>>(h, src, dst, agg);
        k_scale<<<(total4 + T - 1) / T, T, 0, stream>>>(
            (float4*)agg, (float4*)out, dst_norm, src_norm, (l < N_LAYERS - 1) ? 1 : 0);
        float* tmp = h; h = agg; agg = tmp;
    }
}